// EvolutionBank_76836964926222
// MI455X (gfx1250) — compile-verified
//
#include <hip/hip_runtime.h>

#define W 8
#define D 256
#define CAP 16

// ---- CDNA5 async global<->LDS path (guarded so the file always compiles) ----
#if defined(__has_builtin)
#  if __has_builtin(__builtin_amdgcn_global_load_async_to_lds_b128) && \
      __has_builtin(__builtin_amdgcn_global_store_async_from_lds_b128)
#    define USE_ASYNC_LDS 1
#  endif
#endif
#ifndef USE_ASYNC_LDS
#  define USE_ASYNC_LDS 0
#endif

// Builtin signature (from hipcc diagnostics): pointers to 16-byte int vectors,
// global side in addrspace(1), LDS side in addrspace(3), plus two imm ints.
typedef int v4i __attribute__((ext_vector_type(4)));
typedef __attribute__((address_space(1))) v4i* gv4i_p;   // pointer to v4i in global
typedef __attribute__((address_space(3))) v4i* lv4i_p;   // pointer to v4i in LDS

__device__ __forceinline__ void wait_async0() {
#if defined(__has_builtin) && __has_builtin(__builtin_amdgcn_s_wait_asynccnt)
  __builtin_amdgcn_s_wait_asynccnt(0);
#else
  asm volatile("s_wait_asynccnt 0" ::: "memory");
#endif
}

// ---------------- Phase 1: zero per-node counters ----------------
__global__ void __launch_bounds__(256) k_zero(int* __restrict__ cnt, int N) {
  int i = blockIdx.x * blockDim.x + threadIdx.x;
  for (; i < N; i += gridDim.x * blockDim.x) cnt[i] = 0;
}

// ---------------- Phase 2: bucket batch indices per node ----------------
__global__ void __launch_bounds__(256) k_fill(const int* __restrict__ idx,
                                              int* __restrict__ cnt,
                                              int* __restrict__ bucket, int B) {
  int b = blockIdx.x * blockDim.x + threadIdx.x;
  for (; b < B; b += gridDim.x * blockDim.x) {
    int n = idx[b];
    int pos = atomicAdd(&cnt[n], 1);
    if (pos < CAP) bucket[(size_t)n * CAP + pos] = b;
  }
}

// ---- Phase 3: per node, restore batch order (tiny insertion sort) and
//      resolve the winning writer for each of the 8 circular slots.
//      Also emits ptr2 = ptr + count (output part 3). ----
__global__ void __launch_bounds__(256) k_writer(const int* __restrict__ ptr,
                                                const int* __restrict__ cnt,
                                                const int* __restrict__ bucket,
                                                int* __restrict__ writer,
                                                float* __restrict__ out_ptr, int N) {
  int n = blockIdx.x * blockDim.x + threadIdx.x;
  for (; n < N; n += gridDim.x * blockDim.x) {
    int c = cnt[n];
    out_ptr[n] = (float)(ptr[n] + c);
    if (c == 0) continue;
    int cc = (c < CAP) ? c : CAP;
    int bs[CAP];
    for (int j = 0; j < cc; ++j) bs[j] = bucket[(size_t)n * CAP + j];
    for (int j = 1; j < cc; ++j) {            // sort by batch index -> occurrence rank
      int key = bs[j], k2 = j - 1;
      while (k2 >= 0 && bs[k2] > key) { bs[k2 + 1] = bs[k2]; --k2; }
      bs[k2 + 1] = key;
    }
    int p = ptr[n] & 7;
    for (int w = 0; w < W; ++w) {
      int r = (w - p) & 7;                    // occurrences j === r (mod 8) hit slot w
      int win = -1;
      if (r < cc) win = bs[r + (((cc - 1 - r) >> 3) << 3)];  // largest such j wins
      writer[n * W + w] = win;
    }
  }
}

// ---- Phase 4: fused gather. One wave32 per output row (1 KB). Source is
//      either the fresh emb row (if written this batch) or the old bank row.
//      Row copy goes through the CDNA5 async global->LDS->global DMA path. ----
__global__ void __launch_bounds__(256) k_gather(
    const float* __restrict__ bank, const float* __restrict__ times,
    const float* __restrict__ emb, const float* __restrict__ t,
    const int* __restrict__ idx, const int* __restrict__ writer,
    float* __restrict__ out_bank, float* __restrict__ out_times, long long rows) {
#if USE_ASYNC_LDS
  __shared__ __align__(16) float lds[8 * D];
#endif
  const int wave = threadIdx.x >> 5;
  const int lane = threadIdx.x & 31;
  long long row = (long long)blockIdx.x * 8 + wave;
  if (row >= rows) return;

  const int b = (int)(row >> 3);
  const int w = (int)(row & 7);
  const int n = idx[b];
  const int wb = writer[n * W + w];
  const float* src = (wb >= 0) ? (emb + (size_t)wb * D)
                               : (bank + ((size_t)n * W + w) * D);
  float* dst = out_bank + (size_t)row * D;

  if (lane == 0)
    out_times[row] = (wb >= 0) ? t[wb] : times[(size_t)n * W + w];

#if USE_ASYNC_LDS
  float* lb = &lds[wave * D];
  // 32 lanes x 2 x b128 = 1 KB row staged through LDS by the async engine.
  __builtin_amdgcn_global_load_async_to_lds_b128(
      (gv4i_p)(src + lane * 4), (lv4i_p)(lb + lane * 4), 0, 0);
  __builtin_amdgcn_global_load_async_to_lds_b128(
      (gv4i_p)(src + lane * 4 + 128), (lv4i_p)(lb + lane * 4 + 128), 0, 0);
  wait_async0();
  __builtin_amdgcn_global_store_async_from_lds_b128(
      (gv4i_p)(dst + lane * 4), (lv4i_p)(lb + lane * 4), 0, 0);
  __builtin_amdgcn_global_store_async_from_lds_b128(
      (gv4i_p)(dst + lane * 4 + 128), (lv4i_p)(lb + lane * 4 + 128), 0, 0);
  // s_endpgm performs an implicit wait-idle, so outstanding async stores drain.
#else
  const float4* s4 = (const float4*)src;   // global_load_b128
  float4* d4 = (float4*)dst;               // global_store_b128
  d4[lane]      = s4[lane];
  d4[lane + 32] = s4[lane + 32];
#endif
}

extern "C" void kernel_launch(void* const* d_in, const int* in_sizes, int n_in,
                              void* d_out, int out_size, void* d_ws, size_t ws_size,
                              hipStream_t stream) {
  const float* bank  = (const float*)d_in[0];   // (N, 8, 256)
  const float* times = (const float*)d_in[1];   // (N, 8)
  const float* emb   = (const float*)d_in[2];   // (B, 256)
  const float* t     = (const float*)d_in[3];   // (B,)
  const int*   ptr   = (const int*)d_in[4];     // (N,)
  const int*   idx   = (const int*)d_in[5];     // (B,)
  const int N = in_sizes[4];
  const int B = in_sizes[5];

  // workspace: count[N] | bucket[N*CAP] | writer[N*8]  (~20 MB for N=200000)
  int* cnt    = (int*)d_ws;
  int* bucket = cnt + N;
  int* writer = bucket + (size_t)N * CAP;

  float* out_bank  = (float*)d_out;                         // (B, 8, 256)
  float* out_times = out_bank + (size_t)B * W * D;          // (B, 8)
  float* out_ptr   = out_times + (size_t)B * W;             // (N,)

  k_zero<<<(N + 255) / 256, 256, 0, stream>>>(cnt, N);
  k_fill<<<(B + 255) / 256, 256, 0, stream>>>(idx, cnt, bucket, B);
  k_writer<<<(N + 255) / 256, 256, 0, stream>>>(ptr, cnt, bucket, writer, out_ptr, N);

  long long rows = (long long)B * W;
  unsigned gblocks = (unsigned)((rows + 7) / 8);
  k_gather<<<gblocks, 256, 0, stream>>>(bank, times, emb, t, idx, writer,
                                        out_bank, out_times, rows);
}